// CircularMotionEstimationBase_4269197492337
// MI455X (gfx1250) — compile-verified
//
#include <hip/hip_runtime.h>
#include <stdint.h>
#include <math.h>

#define THREADS 256
#define NLM     8192                 // landmarks per batch row
#define EPT     (NLM / THREADS)      // 32 elements per thread
#define TILE    1024                 // elements per async tile (16 KB)
#define NTILES  (NLM / TILE)         // 8
#define TPT     (TILE / THREADS)     // 4 async B128 loads per thread per tile

// ---- LDS layout (dynamic shared) ----
#define SM_XBUF  0                   // [2][1024][4] floats = 32 KB
#define SM_KEYS  32768               // keys[8192] = 32 KB
#define SM_HIST  65536               // hist[256]
#define SM_CNTS  66560               // cnts[256]
#define SM_SH    67584               // sh[4]: {nvalid, rank, prefix key, spare}
#define SM_TOTAL 67600

// order-preserving float->u32 bijection
__device__ __forceinline__ uint32_t f2key(float f) {
  uint32_t u = __float_as_uint(f);
  return (u & 0x80000000u) ? ~u : (u | 0x80000000u);
}

// Full reference-path math, used ONCE per batch row (winner landmark only).
__device__ __forceinline__ float theta_ref(float y2, float y1, float X2, float X1) {
  float r1 = sqrtf(X1 * X1 + y1 * y1);
  float r2 = sqrtf(X2 * X2 + y2 * y2);
  float a1 = atan2f(y1, X1);
  float a2 = atan2f(y2, X2);
  float ratio = r1 / r2;
  return 2.0f * atanf((-sinf(a2) + ratio * sinf(a1)) /
                      (ratio * cosf(a1) + cosf(a2)));
}
__device__ __forceinline__ float curv_ref(float y2, float y1, float X2, float X1,
                                          float th) {
  float r1 = sqrtf(X1 * X1 + y1 * y1);
  float r2 = sqrtf(X2 * X2 + y2 * y2);
  float a1 = atan2f(y1, X1);
  float a2 = atan2f(y2, X2);
  bool stationary = (r1 == r2) && (a1 == a2);
  float radius = r2 * sinf(a1 - a2 - th) /
                 (2.0f * sinf(0.5f * th) * sinf(-a1 + 0.5f * th));
  return stationary ? 0.0f : 1.0f / radius;
}

// Per-wave async copy of this thread's TPT elements of tile `tile` into LDS.
__device__ __forceinline__ void async_tile_load(const float* __restrict__ gbase,
                                                uint32_t lds_buf_base,
                                                int tile, int t) {
#pragma unroll
  for (int q = 0; q < TPT; ++q) {
    int slot = q * THREADS + t;                       // 0..1023, coalesced
    int e = tile * TILE + slot;
    uint64_t gaddr = (uint64_t)(uintptr_t)(gbase + (size_t)e * 4);
    uint32_t laddr = lds_buf_base + (uint32_t)(slot * 16);
    asm volatile("global_load_async_to_lds_b128 %0, %1, off th:TH_LOAD_NT"
                 :: "v"(laddr), "v"(gaddr)
                 : "memory");
  }
}

extern "C" __global__ __launch_bounds__(THREADS)
void cme_median_kernel(const float* __restrict__ x, float* __restrict__ out) {
  extern __shared__ uint8_t smem[];
  float*    xbuf = (float*)(smem + SM_XBUF);
  uint32_t* keys = (uint32_t*)(smem + SM_KEYS);
  uint32_t* hist = (uint32_t*)(smem + SM_HIST);
  uint32_t* cnts = (uint32_t*)(smem + SM_CNTS);
  uint32_t* sh   = (uint32_t*)(smem + SM_SH);

  const int b = blockIdx.x;
  const int t = threadIdx.x;
  const float* xb = x + (size_t)b * NLM * 4;

  if (t < 4) sh[t] = 0;
  __syncthreads();

  const uint32_t xlds = (uint32_t)(uintptr_t)xbuf;  // LDS byte offset

  // ---- Phase 1: async-stream x, key = order-preserving bits of
  //      u=(y1-y2)/(x1+x2); median(theta)==median(u) since theta=2*atan(u). ----
  async_tile_load(xb, xlds + 0, 0, t);
  uint32_t nv = 0;
  for (int j = 0; j < NTILES; ++j) {
    const int buf = j & 1;
    if (j + 1 < NTILES) {
      async_tile_load(xb, xlds + (uint32_t)(((j + 1) & 1) * TILE * 16), j + 1, t);
      asm volatile("s_wait_asynccnt 4" ::: "memory");
    } else {
      asm volatile("s_wait_asynccnt 0" ::: "memory");
    }
#pragma unroll
    for (int q = 0; q < TPT; ++q) {
      const int slot = q * THREADS + t;
      const float4 v = ((const float4*)xbuf)[buf * TILE + slot];
      const float y2 = v.x, y1 = v.y, X2 = v.z, X1 = v.w;
      const bool valid =
          !((y2 == 0.0f) && (y1 == 0.0f) && (X2 == 0.0f) && (X1 == 0.0f));
      const float u = (y1 - y2) / (X1 + X2);          // monotone proxy for theta
      keys[j * TILE + slot] = valid ? f2key(u) : 0xFFFFFFFFu;
      nv += valid ? 1u : 0u;
    }
  }
  atomicAdd(&sh[0], nv);
  __syncthreads();

  const uint32_t nvalid = sh[0];
  if (t == 0) {
    sh[1] = nvalid ? ((nvalid - 1u) >> 1) : 0u;   // lower-median rank
    sh[2] = 0u;
  }

  // ---- Phase 2: 4-pass MSD radix select, parallel scan for bin pick ----
  for (int pass = 0; pass < 4; ++pass) {
    hist[t] = 0;
    __syncthreads();                       // hist zeroed; sh[] visible
    const int shift = 24 - 8 * pass;
    const uint32_t fixedmask =
        (pass == 0) ? 0u : (0xFFFFFFFFu << (32 - 8 * pass));
    const uint32_t prefkey = sh[2];
    const uint32_t kk = sh[1];
#pragma unroll 4
    for (int q = 0; q < EPT; ++q) {
      const uint32_t key = keys[q * THREADS + t];
      if ((key & fixedmask) == prefkey)
        atomicAdd(&hist[(key >> shift) & 0xFFu], 1u);
    }
    __syncthreads();                       // histogram complete
    const uint32_t mycount = hist[t];
    // inclusive Hillis-Steele scan of hist[256]
#pragma unroll
    for (int off = 1; off < THREADS; off <<= 1) {
      const uint32_t add = (t >= off) ? hist[t - off] : 0u;
      __syncthreads();
      hist[t] += add;
      __syncthreads();
    }
    const uint32_t cum = hist[t];
    if (kk < cum && kk >= cum - mycount) { // exactly one thread matches
      sh[1] = kk - (cum - mycount);        // residual rank within bin
      sh[2] = prefkey | ((uint32_t)t << shift);
    }
    __syncthreads();                       // selection visible; hist reusable
  }

  const uint32_t target = sh[2];
  const uint32_t m = sh[1];                // rank among bit-equal keys

  // ---- Phase 3: stable (index-order) tie-break via parallel prefix ----
  uint32_t c = 0;
#pragma unroll 4
  for (int q = 0; q < EPT; ++q) c += (keys[t * EPT + q] == target) ? 1u : 0u;
  cnts[t] = c;
  __syncthreads();
#pragma unroll
  for (int off = 1; off < THREADS; off <<= 1) {
    const uint32_t add = (t >= off) ? cnts[t - off] : 0u;
    __syncthreads();
    cnts[t] += add;
    __syncthreads();
  }
  uint32_t run = cnts[t] - c;              // exclusive prefix
  if (nvalid != 0 && run <= m && m < run + c) {
    for (int q = 0; q < EPT; ++q) {
      const int idx = t * EPT + q;
      if (keys[idx] == target) {
        if (run == m) {
          const float4 v = ((const float4*)xb)[idx];   // 16B reload of winner
          const float th = theta_ref(v.x, v.y, v.z, v.w);
          out[(size_t)b * 2 + 0] = th;
          out[(size_t)b * 2 + 1] = curv_ref(v.x, v.y, v.z, v.w, th);
        }
        ++run;
      }
    }
  }
  if (nvalid == 0 && t == 0) {
    out[(size_t)b * 2 + 0] = 0.0f;
    out[(size_t)b * 2 + 1] = 0.0f;
  }
}

extern "C" void kernel_launch(void* const* d_in, const int* in_sizes, int n_in,
                              void* d_out, int out_size, void* d_ws, size_t ws_size,
                              hipStream_t stream) {
  (void)n_in; (void)out_size; (void)d_ws; (void)ws_size;
  const float* x = (const float*)d_in[0];
  float* out = (float*)d_out;
  const int B = in_sizes[0] / (NLM * 4);  // 2048
  cme_median_kernel<<<dim3(B), dim3(THREADS), SM_TOTAL, stream>>>(x, out);
}